// GCN_41850161332512
// MI455X (gfx1250) — compile-verified
//
#include <hip/hip_runtime.h>
#include <hip/hip_bf16.h>
#include <math.h>

typedef __attribute__((ext_vector_type(2))) float v2f;
typedef __attribute__((ext_vector_type(8))) float v8f;

// ---------------- degree / normalization ----------------

__global__ void k_init_deg(float* __restrict__ deg, int N) {
    int i = blockIdx.x * blockDim.x + threadIdx.x;
    if (i < N) deg[i] = 1.0f;               // self-loop contributes 1
}

__global__ void k_edge_deg(const long long* __restrict__ dst, float* __restrict__ deg, int E) {
    int e = blockIdx.x * blockDim.x + threadIdx.x;
    if (e < E) atomicAdd(&deg[(int)dst[e]], 1.0f);
}

__global__ void k_rsqrt(float* __restrict__ deg, int N) {
    int i = blockIdx.x * blockDim.x + threadIdx.x;
    if (i < N) deg[i] = rsqrtf(deg[i]);     // deg -> dis = deg^-1/2 in place
}

// ---------------- fp32 GEMM via V_WMMA_F32_16X16X4_F32 ----------------
// C[M,NCOL] = A[M,K] @ B[K,NCOL], row-major. One wave per 16-row strip,
// NCOL/16 accumulators per wave (A fragment reused across column tiles).
// K, NCOL compile-time -> fully unrolled WMMA chain.

template<int K, int NCOL>
__global__ void k_gemm_wmma(const float* __restrict__ A, const float* __restrict__ B,
                            float* __restrict__ C, int M) {
    constexpr int TN = NCOL / 16;
    int wave = (int)((blockIdx.x * blockDim.x + threadIdx.x) >> 5);
    int lane = (int)(threadIdx.x & 31);
    int row0 = wave * 16;
    if (row0 >= M) return;                  // wave-uniform: EXEC stays all-ones for WMMA
    int half = lane >> 4;                   // 0 => lanes 0-15, 1 => lanes 16-31
    int r    = lane & 15;

    const float* Arow = A + (size_t)(row0 + r) * K;

    v8f acc[TN] = {};
#pragma unroll
    for (int k = 0; k < K; k += 4) {
        // A fragment: 16x4 f32. VGPR0 = K0 (lo half) / K2 (hi half); VGPR1 = K1 / K3.
        v2f a;
        a.x = Arow[k + 2 * half + 0];
        a.y = Arow[k + 2 * half + 1];
#pragma unroll
        for (int t = 0; t < TN; ++t) {
            // B fragment: 4x16 f32, mirrored layout (rows striped across lanes).
            v2f b;
            b.x = B[(size_t)(k + 2 * half + 0) * NCOL + t * 16 + r];
            b.y = B[(size_t)(k + 2 * half + 1) * NCOL + t * 16 + r];
            acc[t] = __builtin_amdgcn_wmma_f32_16x16x4_f32(
                false, a, false, b, (short)0, acc[t], false, false);
        }
    }
    // C/D layout: VGPR j -> M = j (lanes 0-15) or M = j+8 (lanes 16-31), N = r.
#pragma unroll
    for (int t = 0; t < TN; ++t)
#pragma unroll
        for (int j = 0; j < 8; ++j)
            C[(size_t)(row0 + j + 8 * half) * NCOL + t * 16 + r] = acc[t][j];
}

// ---------------- message passing ----------------

// agg[i] = h[i] * dis[i]^2  (self-loop message; also initializes agg buffer)
__global__ void k_self_msg(const float* __restrict__ h, const float* __restrict__ dis,
                           float* __restrict__ agg, int N, int F4) {
    int t = blockIdx.x * blockDim.x + threadIdx.x;
    int i = t / F4, f = t % F4;
    if (i >= N) return;
    float c = dis[i];
    c = c * c;
    float4 v = ((const float4*)h)[(size_t)i * F4 + f];
    v.x *= c; v.y *= c; v.z *= c; v.w *= c;
    ((float4*)agg)[(size_t)i * F4 + f] = v;
}

// agg[dst] += h[src] * dis[src]*dis[dst], F4 float4-lanes per edge
template<int F4>
__global__ void k_edge_scatter(const long long* __restrict__ src, const long long* __restrict__ dst,
                               const float* __restrict__ dis, const float* __restrict__ h,
                               float* __restrict__ agg, int E) {
    int t = blockIdx.x * blockDim.x + threadIdx.x;
    int e = t / F4;
    int f = (t % F4) * 4;
    if (e >= E) return;
    int s = (int)src[e], d = (int)dst[e];
    float c = dis[s] * dis[d];
    constexpr int F = F4 * 4;
    float4 v = *(const float4*)(h + (size_t)s * F + f);
    float* o = agg + (size_t)d * F + f;
    atomicAdd(o + 0, v.x * c);
    atomicAdd(o + 1, v.y * c);
    atomicAdd(o + 2, v.z * c);
    atomicAdd(o + 3, v.w * c);
}

__global__ void k_bias_relu(float* __restrict__ h, const float* __restrict__ b, int N, int F4) {
    int t = blockIdx.x * blockDim.x + threadIdx.x;
    int i = t / F4, f = t % F4;
    if (i >= N) return;
    float4 v  = ((float4*)h)[(size_t)i * F4 + f];
    float4 bb = ((const float4*)b)[f];
    v.x = fmaxf(v.x + bb.x, 0.0f);
    v.y = fmaxf(v.y + bb.y, 0.0f);
    v.z = fmaxf(v.z + bb.z, 0.0f);
    v.w = fmaxf(v.w + bb.w, 0.0f);
    ((float4*)h)[(size_t)i * F4 + f] = v;
}

__global__ void k_logsoftmax16(const float* __restrict__ agg, const float* __restrict__ b,
                               float* __restrict__ out, int N) {
    int i = blockIdx.x * blockDim.x + threadIdx.x;
    if (i >= N) return;
    float v[16];
    float mx = -INFINITY;
#pragma unroll
    for (int f = 0; f < 16; ++f) {
        v[f] = agg[(size_t)i * 16 + f] + b[f];
        mx = fmaxf(mx, v[f]);
    }
    float s = 0.0f;
#pragma unroll
    for (int f = 0; f < 16; ++f) s += expf(v[f] - mx);
    float lse = mx + logf(s);
#pragma unroll
    for (int f = 0; f < 16; ++f) out[(size_t)i * 16 + f] = v[f] - lse;
}

// ---------------- driver ----------------

extern "C" void kernel_launch(void* const* d_in, const int* in_sizes, int n_in,
                              void* d_out, int out_size, void* d_ws, size_t ws_size,
                              hipStream_t stream) {
    const float*     x  = (const float*)d_in[0];
    const long long* ei = (const long long*)d_in[1];   // int64 edge_index [2,E]
    const float*     W1 = (const float*)d_in[2];
    const float*     b1 = (const float*)d_in[3];
    const float*     W2 = (const float*)d_in[4];
    const float*     b2 = (const float*)d_in[5];

    const int N = in_sizes[0] / 64;
    const int E = in_sizes[1] / 2;
    const long long* src = ei;
    const long long* dst = ei + E;

    float* ws   = (float*)d_ws;
    float* dis  = ws;                          // [N]
    float* bufA = ws + N;                      // [64N]  h1, then h2 + agg2
    float* bufB = bufA + (size_t)N * 64;       // [64N]  agg1 / relu(h1)
    float* h1   = bufA;
    float* agg1 = bufB;
    float* h2   = bufA;                        // reuses dead h1 space [0,16N)
    float* agg2 = bufA + (size_t)N * 16;       // [16N,32N)
    float* out  = (float*)d_out;

    const int B = 256;
    const int waves = (N + 15) / 16;
    const int gemm_blocks = (waves * 32 + B - 1) / B;

    // degrees -> dis = deg^-1/2
    k_init_deg<<<(N + B - 1) / B, B, 0, stream>>>(dis, N);
    k_edge_deg<<<(E + B - 1) / B, B, 0, stream>>>(dst, dis, E);
    k_rsqrt  <<<(N + B - 1) / B, B, 0, stream>>>(dis, N);

    // layer 1: h1 = x @ W1 ; agg1 = D^-1/2 A D^-1/2 h1 ; relu(+b1)
    k_gemm_wmma<64, 64><<<gemm_blocks, B, 0, stream>>>(x, W1, h1, N);
    {
        long tn = (long)N * 16;
        k_self_msg<<<(tn + B - 1) / B, B, 0, stream>>>(h1, dis, agg1, N, 16);
        long te = (long)E * 16;
        k_edge_scatter<16><<<(te + B - 1) / B, B, 0, stream>>>(src, dst, dis, h1, agg1, E);
        k_bias_relu<<<(tn + B - 1) / B, B, 0, stream>>>(agg1, b1, N, 16);
    }

    // layer 2: h2 = relu_h1 @ W2 ; agg2 ; log_softmax(+b2)
    k_gemm_wmma<64, 16><<<gemm_blocks, B, 0, stream>>>(agg1, W2, h2, N);
    {
        long tn = (long)N * 4;
        k_self_msg<<<(tn + B - 1) / B, B, 0, stream>>>(h2, dis, agg2, N, 4);
        long te = (long)E * 4;
        k_edge_scatter<4><<<(te + B - 1) / B, B, 0, stream>>>(src, dst, dis, h2, agg2, E);
        k_logsoftmax16<<<(N + B - 1) / B, B, 0, stream>>>(agg2, b2, out, N);
    }
}